// RelaxedConvolution_58815282151653
// MI455X (gfx1250) — compile-verified
//
#include <hip/hip_runtime.h>
#include <hip/hip_bf16.h>
#include <math.h>

#define N_NODES 50000
#define N_EDGES 800000
#define SILU_C 1.679177f
#define INV_SQRT3 0.57735026918962576f
#define INV_IN  0.25f                 // 1/sqrt(MUL=16)
#define INV_NN  0.25f                 // 1/sqrt(16 neighbors)
#define INV_MID 0.17677669529663687f  // 1/sqrt(32)
#define INV_SQRT_FC 0.125f            // 1/sqrt(64)

typedef __attribute__((ext_vector_type(2))) float v2f;
typedef __attribute__((ext_vector_type(8))) float v8f;
typedef int v4i_ __attribute__((vector_size(16)));   // matches builtin param type

#if defined(__has_builtin)
#if __has_builtin(__builtin_amdgcn_global_load_async_to_lds_b128)
#define HAVE_ASYNC_LDS 1
#endif
#endif

// -------- workspace layout (floats) --------
#define OFF_F0   0
#define OFF_F1   (N_NODES * 16)
#define OFF_SC0  (OFF_F1 + N_NODES * 48)
#define OFF_SC1  (OFF_SC0 + N_NODES * 16)
#define OFF_N0   (OFF_SC1 + N_NODES * 48)
#define OFF_N1   (OFF_N0 + N_NODES * 32)
// total = N_NODES * 256 floats = 51.2 MB

// ---------------- zero accumulators ----------------
__global__ void zero_accum_kernel(float* __restrict__ p, int n) {
    int i = blockIdx.x * blockDim.x + threadIdx.x;
    int stride = gridDim.x * blockDim.x;
    for (; i < n; i += stride) p[i] = 0.0f;
}

// ---------------- node precompute: f0,f1,sc0,sc1 ----------------
__global__ void node_prep_kernel(const float* __restrict__ node_input,
                                 const float* __restrict__ node_attr,
                                 const float* __restrict__ Wsc0,
                                 const float* __restrict__ Wsc1,
                                 const float* __restrict__ Wl10,
                                 const float* __restrict__ Wl11,
                                 float* __restrict__ f0, float* __restrict__ f1,
                                 float* __restrict__ sc0, float* __restrict__ sc1) {
    int t = blockIdx.x * blockDim.x + threadIdx.x;
    int n = t >> 4, wi = t & 15;
    if (n >= N_NODES) return;
    float a = node_attr[n];
    const float* xr = node_input + (size_t)n * 64;
    float s_sc0 = 0.f, s_f0 = 0.f;
    float v_sc1[3] = {0.f, 0.f, 0.f}, v_f1[3] = {0.f, 0.f, 0.f};
#pragma unroll
    for (int u = 0; u < 16; ++u) {
        float x0 = xr[u];
        s_sc0 += x0 * Wsc0[u * 16 + wi];
        s_f0  += x0 * Wl10[u * 16 + wi];
        float wsc1 = Wsc1[u * 16 + wi];
        float wl11 = Wl11[u * 16 + wi];
#pragma unroll
        for (int d = 0; d < 3; ++d) {
            float x1 = xr[16 + u * 3 + d];
            v_sc1[d] += x1 * wsc1;
            v_f1[d]  += x1 * wl11;
        }
    }
    float sa = a * INV_IN;
    sc0[(size_t)n * 16 + wi] = s_sc0 * sa;
    f0 [(size_t)n * 16 + wi] = s_f0  * sa;
#pragma unroll
    for (int d = 0; d < 3; ++d) {
        sc1[(size_t)n * 48 + wi * 3 + d] = v_sc1[d] * sa;
        f1 [(size_t)n * 48 + wi * 3 + d] = v_f1[d]  * sa;
    }
}

// ---------------- edge MLP (WMMA f32) + messages + scatter ----------------
#define TSTRIDE 68  // 64 + 4 pad: bank-conflict-free for row-indexed b64 fragment loads

__global__ __launch_bounds__(256)
void edge_kernel(const float* __restrict__ edge_attr,
                 const float* __restrict__ edge_scalars,
                 const float* __restrict__ fc_w1,
                 const float* __restrict__ fc_w2,
                 const int*   __restrict__ edge_src,
                 const int*   __restrict__ edge_dst,
                 const float* __restrict__ f0,
                 const float* __restrict__ f1,
                 float* __restrict__ n0,
                 float* __restrict__ n1) {
    // weights stored transposed: W[k][n] -> sWt[n*TSTRIDE + k]
    __shared__ __align__(16) float sW1t[64 * TSTRIDE];
    __shared__ __align__(16) float sW2t[64 * TSTRIDE];
    // per-wave staging tile: edge_scalars -> h -> w (reused)
    __shared__ __align__(16) float sTile[8][16 * TSTRIDE];
    // per-wave edge metadata tiles
    __shared__ __align__(16) float sEA[8][64];   // edge_attr, 16 edges x 4
    __shared__ int sSD[8][32];                   // [0..15]=src, [16..31]=dst

    const int tid  = threadIdx.x;
    const int wv   = tid >> 5;
    const int lane = tid & 31;

    for (int i = tid; i < 64 * 64; i += 256) {
        int k = i >> 6, nn = i & 63;
        sW1t[nn * TSTRIDE + k] = fc_w1[i];
        sW2t[nn * TSTRIDE + k] = fc_w2[i];
    }

    const long e0 = (long)blockIdx.x * 128 + (long)wv * 16;
    float* tile = sTile[wv];

    // stage 16x64 edge_scalars tile into LDS
#ifdef HAVE_ASYNC_LDS
#pragma unroll
    for (int i = 0; i < 8; ++i) {
        int lin = i * 128 + lane * 4;
        int r = lin >> 6, c = lin & 63;
        long ge = e0 + r; if (ge >= N_EDGES) ge = N_EDGES - 1;
        __builtin_amdgcn_global_load_async_to_lds_b128(
            (__attribute__((address_space(1))) v4i_*)(edge_scalars + ge * 64 + c),
            (__attribute__((address_space(3))) v4i_*)(tile + r * TSTRIDE + c),
            0, 0);
    }
#else
#pragma unroll
    for (int i = 0; i < 8; ++i) {
        int lin = i * 128 + lane * 4;
        int r = lin >> 6, c = lin & 63;
        long ge = e0 + r; if (ge >= N_EDGES) ge = N_EDGES - 1;
        float4 val = *(const float4*)(edge_scalars + ge * 64 + c);
        *(float4*)(tile + r * TSTRIDE + c) = val;
    }
#endif

    // stage edge_attr (16x4) and src/dst indices for this wave's edges
    {
        long geA = e0 + (lane >> 1); if (geA >= N_EDGES) geA = N_EDGES - 1;
        int c2 = (lane & 1) * 2;
        sEA[wv][(lane >> 1) * 4 + c2 + 0] = edge_attr[geA * 4 + c2 + 0];
        sEA[wv][(lane >> 1) * 4 + c2 + 1] = edge_attr[geA * 4 + c2 + 1];
        long geS = e0 + (lane & 15); if (geS >= N_EDGES) geS = N_EDGES - 1;
        sSD[wv][lane] = (lane < 16) ? edge_src[geS] : edge_dst[geS];
    }

#ifdef HAVE_ASYNC_LDS
#if __has_builtin(__builtin_amdgcn_s_wait_asynccnt)
    __builtin_amdgcn_s_wait_asynccnt(0);
#else
    asm volatile("s_wait_asynccnt 0x0" ::: "memory");
#endif
#endif
    __syncthreads();

    // f32 16x16x4 WMMA fragment indexing (per ISA layout tables)
    const int mrow = lane & 15;          // M (A) / N (B)
    const int koff = (lane >> 4) * 2;    // K sub-offset {0,2}
    const int rbase = (lane < 16) ? 0 : 8;

    v8f acc[4];
#pragma unroll
    for (int nt = 0; nt < 4; ++nt)
        acc[nt] = (v8f){0.f, 0.f, 0.f, 0.f, 0.f, 0.f, 0.f, 0.f};

    // fc1: h(16x64) = ES(16x64) @ W1(64x64)
    for (int k4 = 0; k4 < 64; k4 += 4) {
        v2f a = *(const v2f*)(tile + mrow * TSTRIDE + k4 + koff);
#pragma unroll
        for (int nt = 0; nt < 4; ++nt) {
            v2f b = *(const v2f*)(sW1t + (nt * 16 + mrow) * TSTRIDE + k4 + koff);
            acc[nt] = __builtin_amdgcn_wmma_f32_16x16x4_f32(
                false, a, false, b, (short)0, acc[nt], false, false);
        }
    }
    __syncthreads();

    // silu(h/8)*SILU_C, write h back to tile in A-fragment-friendly layout
#pragma unroll
    for (int nt = 0; nt < 4; ++nt) {
#pragma unroll
        for (int r = 0; r < 8; ++r) {
            float x = acc[nt][r] * INV_SQRT_FC;
            float hv = (x / (1.0f + __expf(-x))) * SILU_C;
            tile[(rbase + r) * TSTRIDE + nt * 16 + mrow] = hv;
        }
    }
    __syncthreads();

#pragma unroll
    for (int nt = 0; nt < 4; ++nt)
        acc[nt] = (v8f){0.f, 0.f, 0.f, 0.f, 0.f, 0.f, 0.f, 0.f};

    // fc2: w(16x64) = h(16x64) @ W2(64x64)
    for (int k4 = 0; k4 < 64; k4 += 4) {
        v2f a = *(const v2f*)(tile + mrow * TSTRIDE + k4 + koff);
#pragma unroll
        for (int nt = 0; nt < 4; ++nt) {
            v2f b = *(const v2f*)(sW2t + (nt * 16 + mrow) * TSTRIDE + k4 + koff);
            acc[nt] = __builtin_amdgcn_wmma_f32_16x16x4_f32(
                false, a, false, b, (short)0, acc[nt], false, false);
        }
    }
    __syncthreads();

#pragma unroll
    for (int nt = 0; nt < 4; ++nt) {
#pragma unroll
        for (int r = 0; r < 8; ++r)
            tile[(rbase + r) * TSTRIDE + nt * 16 + mrow] = acc[nt][r] * INV_SQRT_FC;
    }
    __syncthreads();

    // messages + atomic scatter; lane = (edge-half, channel u)
    const int u = lane & 15;
    const int ehalf = lane >> 4;
    for (int t = 0; t < 8; ++t) {
        int e = ehalf * 8 + t;
        long ge = e0 + e;
        if (ge >= N_EDGES) continue;
        int src = sSD[wv][e];
        int dst = sSD[wv][16 + e];
        float s  = sEA[wv][e * 4 + 0];
        float vx = sEA[wv][e * 4 + 1];
        float vy = sEA[wv][e * 4 + 2];
        float vz = sEA[wv][e * 4 + 3];
        float g0  = f0[(size_t)src * 16 + u];
        float g1x = f1[(size_t)src * 48 + u * 3 + 0];
        float g1y = f1[(size_t)src * 48 + u * 3 + 1];
        float g1z = f1[(size_t)src * 48 + u * 3 + 2];
        float w0 = tile[e * TSTRIDE + u];
        float w1 = tile[e * TSTRIDE + 16 + u];
        float w2 = tile[e * TSTRIDE + 32 + u];
        float w3 = tile[e * TSTRIDE + 48 + u];

        atomicAdd(&n0[(size_t)dst * 32 + u], g0 * s * w0);
        atomicAdd(&n0[(size_t)dst * 32 + 16 + u],
                  (g1x * vx + g1y * vy + g1z * vz) * INV_SQRT3 * w3);
        float gw1 = g0 * w1;
        atomicAdd(&n1[(size_t)dst * 96 + u * 3 + 0], gw1 * vx);
        atomicAdd(&n1[(size_t)dst * 96 + u * 3 + 1], gw1 * vy);
        atomicAdd(&n1[(size_t)dst * 96 + u * 3 + 2], gw1 * vz);
        float sw2 = s * w2;
        atomicAdd(&n1[(size_t)dst * 96 + 48 + u * 3 + 0], g1x * sw2);
        atomicAdd(&n1[(size_t)dst * 96 + 48 + u * 3 + 1], g1y * sw2);
        atomicAdd(&n1[(size_t)dst * 96 + 48 + u * 3 + 2], g1z * sw2);
    }
}

// ---------------- node finalize ----------------
__global__ void node_final_kernel(const float* __restrict__ node_attr,
                                  const float* __restrict__ Wl20,
                                  const float* __restrict__ Wl21,
                                  const float* __restrict__ Wl3,
                                  const float* __restrict__ sc0,
                                  const float* __restrict__ sc1,
                                  const float* __restrict__ n0,
                                  const float* __restrict__ n1,
                                  float* __restrict__ out) {
    int t = blockIdx.x * blockDim.x + threadIdx.x;
    int n = t >> 4, wi = t & 15;
    if (n >= N_NODES) return;
    float a = node_attr[n];
    float o0 = 0.f, ang = 0.f;
    float o1[3] = {0.f, 0.f, 0.f};
#pragma unroll 4
    for (int u = 0; u < 32; ++u) {
        float n0v = n0[(size_t)n * 32 + u];
        o0  += n0v * Wl20[u * 16 + wi];
        ang += n0v * Wl3[u];
        float wl21 = Wl21[u * 16 + wi];
#pragma unroll
        for (int d = 0; d < 3; ++d)
            o1[d] += n1[(size_t)n * 96 + u * 3 + d] * wl21;
    }
    float scale = a * INV_NN * INV_MID;
    o0 *= scale;
    ang *= 0.1f * scale;
    float c = cosf(ang), si = sinf(ang);
    out[(size_t)n * 64 + wi] = c * sc0[(size_t)n * 16 + wi] + si * o0;
#pragma unroll
    for (int d = 0; d < 3; ++d)
        out[(size_t)n * 64 + 16 + wi * 3 + d] =
            c * sc1[(size_t)n * 48 + wi * 3 + d] + si * (o1[d] * scale);
}

// ---------------- launch ----------------
extern "C" void kernel_launch(void* const* d_in, const int* in_sizes, int n_in,
                              void* d_out, int out_size, void* d_ws, size_t ws_size,
                              hipStream_t stream) {
    const float* node_input   = (const float*)d_in[0];
    const float* node_attr    = (const float*)d_in[1];
    const float* edge_attr    = (const float*)d_in[2];
    const float* edge_scalars = (const float*)d_in[3];
    const float* W_sc0        = (const float*)d_in[4];
    const float* W_sc1        = (const float*)d_in[5];
    const float* W_l1_0       = (const float*)d_in[6];
    const float* W_l1_1       = (const float*)d_in[7];
    const float* fc_w1        = (const float*)d_in[8];
    const float* fc_w2        = (const float*)d_in[9];
    const float* W_l2_0       = (const float*)d_in[10];
    const float* W_l2_1       = (const float*)d_in[11];
    const float* W_l3         = (const float*)d_in[12];
    const int*   edge_src     = (const int*)d_in[13];
    const int*   edge_dst     = (const int*)d_in[14];
    float* out = (float*)d_out;

    float* ws  = (float*)d_ws;
    float* f0  = ws + OFF_F0;
    float* f1  = ws + OFF_F1;
    float* sc0 = ws + OFF_SC0;
    float* sc1 = ws + OFF_SC1;
    float* n0  = ws + OFF_N0;
    float* n1  = ws + OFF_N1;

    zero_accum_kernel<<<2048, 256, 0, stream>>>(n0, N_NODES * 128);

    int prep_blocks = (N_NODES * 16 + 255) / 256;
    node_prep_kernel<<<prep_blocks, 256, 0, stream>>>(
        node_input, node_attr, W_sc0, W_sc1, W_l1_0, W_l1_1, f0, f1, sc0, sc1);

    int edge_blocks = (N_EDGES + 127) / 128;
    edge_kernel<<<edge_blocks, 256, 0, stream>>>(
        edge_attr, edge_scalars, fc_w1, fc_w2, edge_src, edge_dst, f0, f1, n0, n1);

    node_final_kernel<<<prep_blocks, 256, 0, stream>>>(
        node_attr, W_l2_0, W_l2_1, W_l3, sc0, sc1, n0, n1, out);
}